// TransMLACruncher_47304769798935
// MI455X (gfx1250) — compile-verified
//
#include <hip/hip_runtime.h>
#include <stdint.h>

// ---- problem constants ----
#define Bsz   4
#define Hn    8
#define Sn    8192
#define Dn    128
#define HALF  64
#define RANK  512
#define INF_K 512            // K-derived features
#define INF_T 1536           // total in-features
#define TOK   (Bsz * Sn)     // 32768 tokens
#define NCHUNK (INF_T / 32)  // 48 k-chunks

typedef __attribute__((ext_vector_type(2))) float v2f;
typedef __attribute__((ext_vector_type(4))) float v4f;
typedef __attribute__((ext_vector_type(8))) float v8f;

typedef __attribute__((ext_vector_type(4))) unsigned u32x4;
typedef __attribute__((ext_vector_type(8))) int      i32x8;
typedef __attribute__((ext_vector_type(4))) int      i32x4;

// -------------------------------------------------------------------------
// gfx1250 Tensor Data Mover: 2D tile (global -> LDS), hardware row padding.
//   td0       : tile width in elements (f32), contiguous in memory
//   td1       : number of rows
//   stride    : row stride in elements
//   padi/pada : D# pad_interval / pad_amount codes (LDS row padding)
// Descriptor packed per CDNA5 ISA 8.3/8.4 (count=1, type=2, data_size=4B).
// Issued by one wave; tracked by TENSORcnt. (clang-23 6-arg builtin form.)
// -------------------------------------------------------------------------
__device__ __forceinline__ void tdm_load_2d(unsigned lds_off, const void* g,
                                            unsigned td0, unsigned td1,
                                            unsigned stride, unsigned padi,
                                            unsigned pada) {
  uint64_t ga = (uint64_t)(uintptr_t)g;
  u32x4 g0;
  g0[0] = 1u;                                              // count=1, user mode
  g0[1] = lds_off;                                         // lds_addr (bytes)
  g0[2] = (unsigned)ga;                                    // global_addr[31:0]
  g0[3] = (unsigned)((ga >> 32) & 0x1FFFFFFu) | (2u << 30);// addr[56:32], type=2
  i32x8 g1;
  g1[0] = (int)((2u << 16) | (1u << 20) | (padi << 22) | (pada << 25));
  g1[1] = (int)((td0 & 0xFFFFu) << 16);   // tensor_dim0[15:0]
  g1[2] = (int)((td1 & 0xFFFFu) << 16);   // tensor_dim0 hi=0 | tensor_dim1[15:0]
  g1[3] = (int)((td0 & 0xFFFFu) << 16);   // tensor_dim1 hi=0 | tile_dim0
  g1[4] = (int)(td1 & 0xFFFFu);           // tile_dim1 | tile_dim2=0
  g1[5] = (int)stride;                    // tensor_dim0_stride[31:0]
  g1[6] = 0;                              // stride hi | dim1_stride lo
  g1[7] = 0;
  i32x4 gz4 = {0, 0, 0, 0};
  i32x8 gz8 = {0, 0, 0, 0, 0, 0, 0, 0};
  __builtin_amdgcn_tensor_load_to_lds(g0, g1, gz4, gz4, gz8, 0);
}

// -------------------------------------------------------------------------
// Kernel 0: fused weight  W[1536][512] in workspace.
//   rows 0..511:   W[h*64+d, r] = (1/alpha) * sum_e U_l[e,d] * R_KV[h*64+e, r]
//   rows 512..1535: copy of R_KV
// -------------------------------------------------------------------------
__global__ __launch_bounds__(256) void build_w_kernel(
    const float* __restrict__ Ul, const float* __restrict__ Rkv,
    const float* __restrict__ alpha, float* __restrict__ W) {
  int idx = blockIdx.x * 256 + threadIdx.x;          // 0 .. 1536*512-1
  int j = idx >> 9;                                  // feature row
  int r = idx & (RANK - 1);                          // rank col
  if (j < INF_K) {
    float inva = 1.0f / alpha[0];
    int h = j >> 6, d = j & 63;
    float acc = 0.0f;
#pragma unroll 8
    for (int e = 0; e < HALF; ++e)
      acc += Ul[e * HALF + d] * Rkv[(size_t)(h * HALF + e) * RANK + r];
    W[idx] = acc * inva;
  } else {
    W[idx] = Rkv[idx];
  }
}

// -------------------------------------------------------------------------
// Kernel 1: C_kv = X(32768 x 1536) @ W(1536 x 512)
//   X[t, k]  (k < 512)  = K[b, k/64, s, 64 + k%64]          (t = b*S + s)
//   X[t, k]  (k >= 512) = V[b, (k-512)/128, s, (k-512)%128]
// Block = 256 thr = 8 waves; C block-tile 256x128; K-chunk 32; double-buffered
// LDS panels: A staged by TDM (hardware padded stride 36), B staged through
// registers (transpose) into [n][k] stride 34. Wave w: rows [w*32,+32),
// 2 m-subtiles x 8 n-tiles = 16 v8f accumulators.
// -------------------------------------------------------------------------
#define ALDS_STRIDE 36   // 32 k-floats + 4 pad  (TDM: padi=4 -> 32dw, pada=3 -> 4dw)
#define BT_STRIDE   34   // 32 k-floats + pad (even keeps b64 align)

__global__ __launch_bounds__(256) void gemm_ckv_kernel(
    const float* __restrict__ Kp, const float* __restrict__ Vp,
    const float* __restrict__ W, float* __restrict__ Ckv) {
  __shared__ float Alds[2][256 * ALDS_STRIDE];   // [row][k], hw-padded
  __shared__ float Bt[2][128 * BT_STRIDE];       // [n][k], transposed W

  const int tid  = threadIdx.x;
  const int wave = tid >> 5;
  const int lane = tid & 31;
  const int l16  = lane & 15;
  const int hi   = lane >> 4;                 // 0: K0/K1 half, 1: K2/K3 half

  const long tokenBase = (long)blockIdx.x * 256;
  const int  b0 = (int)(tokenBase >> 13);     // 256 | 8192 -> single batch
  const int  s0 = (int)(tokenBase & (Sn - 1));
  const int  ncol0 = blockIdx.y * 128;

  // ---- stage A chunk via TDM (wave 0 only; EXEC ignored by TDM) ----
  auto stage_A = [&](int buf, int k0) {
    const float* base;
    if (k0 < INF_K) {                         // K_y features
      int h = k0 >> 6, d = k0 & 63;
      base = Kp + ((((long)b0 * Hn + h) * Sn + s0) * Dn + HALF + d);
    } else {                                  // V features
      int j = k0 - INF_K;
      int h = j >> 7, d = j & 127;
      base = Vp + ((((long)b0 * Hn + h) * Sn + s0) * Dn + d);
    }
    tdm_load_2d((unsigned)(uintptr_t)&Alds[buf][0], base,
                /*td0=*/32, /*td1=*/256, /*stride=*/Dn, /*padi=*/4, /*pada=*/3);
  };

  // ---- stage B chunk transposed: W[k0+kk, ncol0+n] -> Bt[buf][n][kk] ----
  auto stage_B = [&](int buf, int k0) {
#pragma unroll
    for (int it = 0; it < 4; ++it) {
      int idx = it * 256 + tid;               // 0..1023
      int kk  = idx >> 5;                     // 0..31
      int n4  = idx & 31;                     // float4 col slot
      v4f w = *(const v4f*)&W[(size_t)(k0 + kk) * RANK + ncol0 + n4 * 4];
#pragma unroll
      for (int j = 0; j < 4; ++j)
        Bt[buf][(n4 * 4 + j) * BT_STRIDE + kk] = w[j];
    }
  };

  v8f acc[2][8] = {};

  // prologue: stage chunk 0
  if (wave == 0) stage_A(0, 0);
  stage_B(0, 0);

  for (int c = 0; c < NCHUNK; ++c) {
    const int cur = c & 1;
    if (c + 1 < NCHUNK) {
      const int k0n = (c + 1) * 32;
      if (wave == 0) stage_A(cur ^ 1, k0n);   // DMA overlaps compute of c
      stage_B(cur ^ 1, k0n);
      if (k0n + 32 < INF_T)                   // L2 prefetch of next W panel
        __builtin_prefetch(&W[(size_t)(k0n + 32) * RANK + ncol0] + tid * 2, 0, 1);
      __builtin_amdgcn_s_wait_tensorcnt(1);   // in-order: chunk c has landed
    } else {
      __builtin_amdgcn_s_wait_tensorcnt(0);
    }
    __syncthreads();

    // ---- compute: 8 k-steps x (2 m-subtiles x 8 n-tiles) WMMAs ----
    const int rowb = wave * 32 + l16;
#pragma unroll
    for (int ks = 0; ks < 8; ++ks) {
      v2f a0 = *(const v2f*)&Alds[cur][(rowb +  0) * ALDS_STRIDE + ks * 4 + hi * 2];
      v2f a1 = *(const v2f*)&Alds[cur][(rowb + 16) * ALDS_STRIDE + ks * 4 + hi * 2];
#pragma unroll
      for (int t = 0; t < 8; ++t) {
        v2f bfrag = *(const v2f*)&Bt[cur][(t * 16 + l16) * BT_STRIDE + ks * 4 + hi * 2];
        acc[0][t] = __builtin_amdgcn_wmma_f32_16x16x4_f32(
            false, a0, false, bfrag, (short)0, acc[0][t], false, false);
        acc[1][t] = __builtin_amdgcn_wmma_f32_16x16x4_f32(
            false, a1, false, bfrag, (short)0, acc[1][t], false, false);
      }
    }
    __syncthreads();
  }

  // ---- store C tile ----
#pragma unroll
  for (int m = 0; m < 2; ++m) {
    const long rowStart = tokenBase + wave * 32 + m * 16;
#pragma unroll
    for (int t = 0; t < 8; ++t) {
#pragma unroll
      for (int v = 0; v < 8; ++v) {
        long row = rowStart + v + hi * 8;
        int  col = ncol0 + t * 16 + l16;
        Ckv[row * RANK + col] = acc[m][t][v];
      }
    }
  }
}

// -------------------------------------------------------------------------
// Kernel 2: K_rope[row, e] = sum_d K[row*128 + d] * U_l[e, d]  (d,e < 64)
// Block = 256 thr = 8 waves; 128 rows per block; K rows and U_l staged in
// LDS by TDM (hardware padded stride 68); each wave: 16 rows x 64 cols.
// -------------------------------------------------------------------------
#define ULDS_STRIDE 68   // 64 + 4 pad  (TDM: padi=5 -> 64dw, pada=3 -> 4dw)
#define KLDS_STRIDE 68

__global__ __launch_bounds__(256) void rope_kernel(
    const float* __restrict__ Kp, const float* __restrict__ Ul,
    float* __restrict__ Krope) {
  __shared__ float Ulds[HALF * ULDS_STRIDE];   // [e][d]
  __shared__ float Klds[128 * KLDS_STRIDE];    // [row][d], d < 64

  const int tid  = threadIdx.x;
  const int wave = tid >> 5;
  const int lane = tid & 31;
  const int l16  = lane & 15;
  const int hi   = lane >> 4;

  const long blockRow0 = (long)blockIdx.x * 128;

  if (wave == 0) {
    // U_l: 64x64 tile, contiguous rows (stride 64)
    tdm_load_2d((unsigned)(uintptr_t)&Ulds[0], Ul,
                /*td0=*/64, /*td1=*/64, /*stride=*/HALF, /*padi=*/5, /*pada=*/3);
    // K first halves: 128 rows x 64 d, row stride 128
    tdm_load_2d((unsigned)(uintptr_t)&Klds[0], Kp + blockRow0 * Dn,
                /*td0=*/64, /*td1=*/128, /*stride=*/Dn, /*padi=*/5, /*pada=*/3);
  }
  __builtin_amdgcn_s_wait_tensorcnt(0);
  __syncthreads();

  v8f acc[4] = {};

#pragma unroll
  for (int ks = 0; ks < 16; ++ks) {
    v2f afrag = *(const v2f*)&Klds[(wave * 16 + l16) * KLDS_STRIDE + ks * 4 + hi * 2];
#pragma unroll
    for (int t = 0; t < 4; ++t) {
      int nn = t * 16 + l16;                  // output col e
      // B[k, n] = U_l[n, k]
      v2f bfrag = *(const v2f*)&Ulds[nn * ULDS_STRIDE + ks * 4 + hi * 2];
      acc[t] = __builtin_amdgcn_wmma_f32_16x16x4_f32(
          false, afrag, false, bfrag, (short)0, acc[t], false, false);
    }
  }

  const long rowBase = blockRow0 + wave * 16;
#pragma unroll
  for (int t = 0; t < 4; ++t) {
#pragma unroll
    for (int v = 0; v < 8; ++v) {
      long row = rowBase + v + hi * 8;
      Krope[row * HALF + t * 16 + l16] = acc[t][v];
    }
  }
}

// -------------------------------------------------------------------------
extern "C" void kernel_launch(void* const* d_in, const int* in_sizes, int n_in,
                              void* d_out, int out_size, void* d_ws, size_t ws_size,
                              hipStream_t stream) {
  const float* Kp    = (const float*)d_in[0];  // (B,H,S,D)
  const float* Vp    = (const float*)d_in[1];  // (B,H,S,D)
  const float* Ul    = (const float*)d_in[2];  // (64,64)
  const float* Rkv   = (const float*)d_in[3];  // (1536,512)
  const float* alpha = (const float*)d_in[4];  // (1,)

  float* out   = (float*)d_out;
  float* Ckv   = out;                                   // (B,S,512)
  float* Krope = out + (size_t)Bsz * Sn * RANK;         // (B,H,S,64)
  float* W     = (float*)d_ws;                          // 1536*512 f32 = 3 MiB

  // 1) fused weight
  build_w_kernel<<<(INF_T * RANK) / 256, 256, 0, stream>>>(Ul, Rkv, alpha, W);

  // 2) main GEMM: grid (M/256, N/128) = (128, 4)
  gemm_ckv_kernel<<<dim3(TOK / 256, RANK / 128), 256, 0, stream>>>(Kp, Vp, W, Ckv);

  // 3) rope rotation GEMM: 262144 rows / 128 per block
  rope_kernel<<<(Bsz * Hn * Sn) / 128, 256, 0, stream>>>(Kp, Ul, Krope);
}